// SimpleMamba_57432302682153
// MI455X (gfx1250) — compile-verified
//
#include <hip/hip_runtime.h>

#define BSZ 8
#define SEQ 2048
#define DM 256
#define NST 16
#define IN_FEAT 512
#define IN_OFF 128
#define LDSX 260      // padded row stride (floats): 16B aligned, bank-conflict free
#define CHUNK 64      // scan chunk length (power of two; A^CHUNK via squaring)
#define NCHUNK (SEQ / CHUNK)   // 32

typedef __attribute__((ext_vector_type(2))) float v2f;
typedef __attribute__((ext_vector_type(8))) float v8f;
typedef __attribute__((ext_vector_type(4))) int   v4i;

// D(16x16,f32) = A(16x4,f32) x B(4x16,f32) + C   — CDNA5 V_WMMA_F32_16X16X4_F32
__device__ __forceinline__ v8f wmma4_f32(v2f a, v2f b, v8f c) {
    return __builtin_amdgcn_wmma_f32_16x16x4_f32(
        false, a, false, b, (short)0, c, false, false);
}

// ---- CDNA5 async global->LDS staging (ASYNCcnt path), with safe fallback ----
#if defined(__has_builtin)
# if __has_builtin(__builtin_amdgcn_global_load_async_to_lds_b128)
#  define HAVE_ASYNC_LDS 1
# endif
# if __has_builtin(__builtin_amdgcn_tensor_load_to_lds)
#  define HAVE_TDM 1
# endif
#endif

#ifdef HAVE_ASYNC_LDS
typedef __attribute__((address_space(1))) v4i as1_v4i;   // global
typedef __attribute__((address_space(3))) v4i as3_v4i;   // LDS
#endif

__device__ __forceinline__ void stage_b128(const float* g, float* l) {
#ifdef HAVE_ASYNC_LDS
    // global ptr: bit pattern of a generic pointer to global memory == AS1 ptr,
    // so an integer round-trip is exact (and drops const). LDS ptr: explicit
    // generic->AS3 address space conversion.
    __builtin_amdgcn_global_load_async_to_lds_b128(
        (as1_v4i*)(unsigned long long)g,
        (as3_v4i*)l,
        /*offset=*/0, /*cpol=*/0);
#else
    *(float4*)l = *(const float4*)g;
#endif
}

__device__ __forceinline__ void stage_wait() {
#ifdef HAVE_ASYNC_LDS
# if __has_builtin(__builtin_amdgcn_s_wait_asynccnt)
    __builtin_amdgcn_s_wait_asynccnt(0);
# else
    asm volatile("s_wait_asynccnt 0x0" ::: "memory");
# endif
#endif
}

// ---- CDNA5 Tensor Data Mover: DMA a 2D f32 tile global->LDS (TENSORcnt) ----
// D# per CDNA5 ISA. The TDM pad unit inserts
// 4 DWORDs after every 256 DWORDs stored, producing the 260-float padded LDS
// row stride in hardware (pad_interval=7 -> 256 DW, pad_amount=3 -> 4 DW).
#ifdef HAVE_TDM
typedef __attribute__((ext_vector_type(4))) unsigned int v4u;
typedef __attribute__((ext_vector_type(8))) int          v8i;

__device__ __forceinline__ void tdm_load_tile_2d(const float* gsrc, float* lds_dst,
                                                 unsigned rows, unsigned row_elems,
                                                 unsigned src_stride_elems) {
    unsigned long long ga = (unsigned long long)gsrc;
    unsigned lds_addr =
        (unsigned)(unsigned long long)(__attribute__((address_space(3))) float*)lds_dst;

    v4u g0;
    g0[0] = 1u;                                        // count=1, user descriptor
    g0[1] = lds_addr;                                  // LDS byte address
    g0[2] = (unsigned)(ga & 0xFFFFFFFFu);              // global_addr[31:0]
    g0[3] = (unsigned)((ga >> 32) & 0x1FFFFFFu)        // global_addr[56:32]
          | (2u << 30);                                // type = 2 ("image")

    const unsigned td0 = src_stride_elems;             // tensor_dim0 (covers tile x)
    const unsigned td1 = rows;                         // tensor_dim1
    v8i g1;
    g1[0] = (int)((2u << 16)                           // data_size = 4 bytes
                | (1u << 20)                           // pad_enable
                | (7u << 22)                           // pad_interval = 256 DWORDs
                | (3u << 25));                         // pad_amount   = 4 DWORDs
    g1[1] = (int)((td0 & 0xFFFFu) << 16);              // tensor_dim0[15:0] @ bits 63:48
    g1[2] = (int)((td0 >> 16) | ((td1 & 0xFFFFu) << 16)); // td0 hi | td1 lo
    g1[3] = (int)((td1 >> 16) | (row_elems << 16));    // td1 hi | tile_dim0
    g1[4] = (int)rows;                                 // tile_dim1 (tile_dim2 = 0)
    g1[5] = (int)src_stride_elems;                     // tensor_dim0_stride[31:0]
    g1[6] = 0;                                         // stride0 hi | stride1 lo
    g1[7] = 0;
    v4i zz4 = {0, 0, 0, 0};                            // groups 2/3 unused (2D tensor)
    v8i zz8 = {0, 0, 0, 0, 0, 0, 0, 0};
    __builtin_amdgcn_tensor_load_to_lds(g0, g1, zz4, zz4, zz8, /*cpol=*/0);
}
#endif

// ---------------------------------------------------------------------------
// Kernel 1: fused in_proj GEMM + gate GEMM + sigmoid gating.
// 512 blocks (one per (batch, 32-token tile)), 256 threads = 8 waves.
// Each wave owns a 32x32 output tile = 4 WMMA accumulator chains; B fragments
// are reused across both M sub-tiles (2 A-loads + 2 B-loads -> 4 WMMAs).
// Input tile is staged by the Tensor Data Mover (one DMA per block).
// ---------------------------------------------------------------------------
__global__ __launch_bounds__(256)
void k_proj_gate(const float* __restrict__ xin,
                 const float* __restrict__ in_w,  const float* __restrict__ in_b,
                 const float* __restrict__ gate_w,const float* __restrict__ gate_b,
                 float* __restrict__ xproj, float* __restrict__ xg) {
    __shared__ float lds_x[32 * LDSX];
    __shared__ float lds_p[32 * LDSX];

    const int blk = blockIdx.x;
    const int b   = blk >> 6;          // 64 tiles per batch
    const int s0  = (blk & 63) << 5;   // 32-token tile start
    const int t   = threadIdx.x;

    // Stage the input slice [32 tokens x 256 feats] (cols 128..383) into LDS.
    {
        const float* src = xin + ((size_t)(b * SEQ + s0)) * IN_FEAT + IN_OFF;
#ifdef HAVE_TDM
        if (t < 32) {                                  // wave 0 issues the DMA
            tdm_load_tile_2d(src, lds_x, 32, DM, IN_FEAT);
            __builtin_amdgcn_s_wait_tensorcnt(0);
        }
#else
        #pragma unroll
        for (int e = 0; e < 8; ++e) {
            int idx = t + e * 256;            // 2048 float4 chunks total
            int row = idx >> 6, c4 = idx & 63;
            stage_b128(src + (size_t)row * IN_FEAT + c4 * 4,
                       &lds_x[row * LDSX + c4 * 4]);
        }
        stage_wait();
#endif
    }
    __syncthreads();

    const int lane = t & 31;
    const int wave = t >> 5;
    const int r    = lane & 15;   // A: token row / B: weight row offset
    const int hi   = lane >> 4;   // 0 -> holds k,k+1 ; 1 -> holds k+2,k+3
    const int n0   = wave << 5;   // wave owns N columns [n0, n0+32)

    // ---------------- GEMM1: xproj = x @ in_w^T ----------------
    v8f a00 = {}, a01 = {}, a10 = {}, a11 = {};   // [m-subtile][n-subtile]
    for (int k = 0; k < DM; k += 4) {
        const int kk = k + 2 * hi;
        v2f fa0 = *(const v2f*)&lds_x[r * LDSX + kk];
        v2f fa1 = *(const v2f*)&lds_x[(16 + r) * LDSX + kk];
        const float* w0 = in_w + (size_t)(n0 + r) * DM + kk;
        const float* w1 = w0 + (size_t)16 * DM;
        v2f fb0; fb0.x = w0[0]; fb0.y = w0[1];
        v2f fb1; fb1.x = w1[0]; fb1.y = w1[1];
        a00 = wmma4_f32(fa0, fb0, a00);
        a10 = wmma4_f32(fa1, fb0, a10);
        a01 = wmma4_f32(fa0, fb1, a01);
        a11 = wmma4_f32(fa1, fb1, a11);
    }

    float* gxp = xproj + ((size_t)(b * SEQ + s0)) * DM;
    {
        const int n = n0 + r, n2 = n + 16;
        const float bi = in_b[n], bi2 = in_b[n2];
        #pragma unroll
        for (int rr = 0; rr < 8; ++rr) {
            int m = rr + 8 * hi;   // C/D layout: VGPR rr -> row m (m+8 for hi lanes)
            float v;
            v = a00[rr] + bi;  lds_p[m * LDSX + n ] = v;        gxp[(size_t)m * DM + n ] = v;
            v = a01[rr] + bi2; lds_p[m * LDSX + n2] = v;        gxp[(size_t)m * DM + n2] = v;
            v = a10[rr] + bi;  lds_p[(m+16) * LDSX + n ] = v;   gxp[(size_t)(m+16) * DM + n ] = v;
            v = a11[rr] + bi2; lds_p[(m+16) * LDSX + n2] = v;   gxp[(size_t)(m+16) * DM + n2] = v;
        }
    }
    __syncthreads();

    // ---------------- GEMM2: gate = sigmoid(xproj @ gate_w^T) ----------------
    v8f g00 = {}, g01 = {}, g10 = {}, g11 = {};
    for (int k = 0; k < DM; k += 4) {
        const int kk = k + 2 * hi;
        v2f fa0 = *(const v2f*)&lds_p[r * LDSX + kk];
        v2f fa1 = *(const v2f*)&lds_p[(16 + r) * LDSX + kk];
        const float* w0 = gate_w + (size_t)(n0 + r) * DM + kk;
        const float* w1 = w0 + (size_t)16 * DM;
        v2f fb0; fb0.x = w0[0]; fb0.y = w0[1];
        v2f fb1; fb1.x = w1[0]; fb1.y = w1[1];
        g00 = wmma4_f32(fa0, fb0, g00);
        g10 = wmma4_f32(fa1, fb0, g10);
        g01 = wmma4_f32(fa0, fb1, g01);
        g11 = wmma4_f32(fa1, fb1, g11);
    }

    float* gxg = xg + ((size_t)(b * SEQ + s0)) * DM;
    {
        const int n = n0 + r, n2 = n + 16;
        const float gb = gate_b[n], gb2 = gate_b[n2];
        #pragma unroll
        for (int rr = 0; rr < 8; ++rr) {
            int m = rr + 8 * hi;
            float s;
            s = 1.0f / (1.0f + __expf(-(g00[rr] + gb )));
            gxg[(size_t)m * DM + n ] = lds_p[m * LDSX + n ] * s;
            s = 1.0f / (1.0f + __expf(-(g01[rr] + gb2)));
            gxg[(size_t)m * DM + n2] = lds_p[m * LDSX + n2] * s;
            s = 1.0f / (1.0f + __expf(-(g10[rr] + gb )));
            gxg[(size_t)(m+16) * DM + n ] = lds_p[(m+16) * LDSX + n ] * s;
            s = 1.0f / (1.0f + __expf(-(g11[rr] + gb2)));
            gxg[(size_t)(m+16) * DM + n2] = lds_p[(m+16) * LDSX + n2] * s;
        }
    }
}

// ---------------------------------------------------------------------------
// Chunked parallel scan: h_s = A ⊙ h_{s-1} + xg_s ⊙ B is elementwise-linear,
// so split S=2048 into 32 chunks of 64.
// Pass 1: per-(b,chunk,d) local scan with h0 = 0 -> chunk contribution v_c.
// Pass 2: sequential combine over 32 chunks: hstart_c = A^64 ⊙ hstart_{c-1} + v_{c-1}
// Pass 3: re-run each chunk from hstart_c, emit y + depthwise conv, in place.
// Serial depth drops 2048 -> 64 (+32), parallelism rises 2K -> 65K threads.
// ---------------------------------------------------------------------------
__global__ __launch_bounds__(256)
void k_scan_local(const float* __restrict__ xg, const float* __restrict__ A_log,
                  const float* __restrict__ Bs, float* __restrict__ cs) {
    const int d = threadIdx.x;
    const int b = blockIdx.x >> 5;       // NCHUNK = 32
    const int c = blockIdx.x & 31;

    float Aa[NST], Bb[NST], h[NST];
    #pragma unroll
    for (int n = 0; n < NST; ++n) {
        Aa[n] = -__expf(A_log[d * NST + n]);
        Bb[n] = Bs[d * NST + n];
        h[n]  = 0.f;
    }
    const float* x = xg + ((size_t)(b * SEQ + c * CHUNK)) * DM + d;
    for (int s = 0; s < CHUNK; s += 8) {
        float xt[8];
        #pragma unroll
        for (int j = 0; j < 8; ++j) xt[j] = x[(size_t)(s + j) * DM];
        #pragma unroll
        for (int j = 0; j < 8; ++j) {
            #pragma unroll
            for (int n = 0; n < NST; ++n)
                h[n] = fmaf(Aa[n], h[n], xt[j] * Bb[n]);
        }
    }
    float* slot = cs + (((size_t)b * NCHUNK + c) * DM + d) * NST;
    #pragma unroll
    for (int n = 0; n < NST; ++n) slot[n] = h[n];
}

__global__ __launch_bounds__(256)
void k_scan_combine(const float* __restrict__ A_log, float* __restrict__ cs) {
    const int d = threadIdx.x;
    const int b = blockIdx.x;

    float AL[NST];
    #pragma unroll
    for (int n = 0; n < NST; ++n) {
        float A = -__expf(A_log[d * NST + n]);
        float p = A;                      // A^(2^6) = A^64 = A^CHUNK
        #pragma unroll
        for (int q = 0; q < 6; ++q) p = p * p;
        AL[n] = p;
    }
    float* base = cs + ((size_t)b * NCHUNK * DM + (size_t)d) * NST;
    float hprev[NST], vprev[NST];
    #pragma unroll
    for (int n = 0; n < NST; ++n) {       // chunk 0: hstart = 0, remember v0
        vprev[n] = base[n];
        hprev[n] = 0.f;
        base[n]  = 0.f;
    }
    for (int c = 1; c < NCHUNK; ++c) {
        float* slot = base + (size_t)c * DM * NST;
        float vcur[NST];
        #pragma unroll
        for (int n = 0; n < NST; ++n) vcur[n] = slot[n];
        #pragma unroll
        for (int n = 0; n < NST; ++n) {
            hprev[n] = fmaf(AL[n], hprev[n], vprev[n]);
            slot[n]  = hprev[n];
            vprev[n] = vcur[n];
        }
    }
}

__global__ __launch_bounds__(256)
void k_scan_final(const float* __restrict__ xproj, float* __restrict__ xy,
                  const float* __restrict__ A_log, const float* __restrict__ Bs,
                  const float* __restrict__ Cs,    const float* __restrict__ Ds,
                  const float* __restrict__ conv_w, const float* __restrict__ conv_b,
                  const float* __restrict__ cs) {
    const int d = threadIdx.x;
    const int b = blockIdx.x >> 5;
    const int c = blockIdx.x & 31;
    const int s0 = c * CHUNK;

    float Aa[NST], Bb[NST], Cc[NST], h[NST];
    const float* slot = cs + (((size_t)b * NCHUNK + c) * DM + d) * NST;
    #pragma unroll
    for (int n = 0; n < NST; ++n) {
        Aa[n] = -__expf(A_log[d * NST + n]);
        Bb[n] = Bs[d * NST + n];
        Cc[n] = Cs[d * NST + n];
        h[n]  = slot[n];                  // correct start state for this chunk
    }
    const float Dd = Ds[d];
    const float w0 = conv_w[d * 3 + 0], w1 = conv_w[d * 3 + 1], w2 = conv_w[d * 3 + 2];
    const float cb = conv_b[d];

    const float* xp = xproj + (size_t)b * SEQ * DM + d;
    float*       yx = xy    + (size_t)b * SEQ * DM + d;

    float xm1 = (s0 == 0) ? 0.f : xp[(size_t)(s0 - 1) * DM];
    float x0  = xp[(size_t)s0 * DM];

    for (int s = s0; s < s0 + CHUNK; s += 8) {
        float xt[8], xn[8];
        #pragma unroll
        for (int j = 0; j < 8; ++j) {             // issue all loads up front
            xt[j] = yx[(size_t)(s + j) * DM];
            int sp = s + j + 1;
            xn[j] = (sp < SEQ) ? xp[(size_t)sp * DM] : 0.f;
        }
        #pragma unroll
        for (int j = 0; j < 8; ++j) {
            const float x = xt[j];
            float y0 = Dd * x, y1 = 0.f, y2 = 0.f, y3 = 0.f;
            #pragma unroll
            for (int n = 0; n < NST; n += 4) {
                h[n+0] = fmaf(Aa[n+0], h[n+0], x * Bb[n+0]);
                h[n+1] = fmaf(Aa[n+1], h[n+1], x * Bb[n+1]);
                h[n+2] = fmaf(Aa[n+2], h[n+2], x * Bb[n+2]);
                h[n+3] = fmaf(Aa[n+3], h[n+3], x * Bb[n+3]);
                y0 = fmaf(h[n+0], Cc[n+0], y0);
                y1 = fmaf(h[n+1], Cc[n+1], y1);
                y2 = fmaf(h[n+2], Cc[n+2], y2);
                y3 = fmaf(h[n+3], Cc[n+3], y3);
            }
            float xc = fmaf(w0, xm1, fmaf(w1, x0, fmaf(w2, xn[j], cb)));
            yx[(size_t)(s + j) * DM] = (y0 + y1) + (y2 + y3) + xc;
            xm1 = x0; x0 = xn[j];
        }
    }
}

// ---------------------------------------------------------------------------
// Kernel 3: out_proj GEMM: out = ytot @ out_w^T + out_b  (32-token M tiles).
// Tile staged via async global->LDS (ASYNCcnt path).
// ---------------------------------------------------------------------------
__global__ __launch_bounds__(256)
void k_outproj(const float* __restrict__ ytot,
               const float* __restrict__ out_w, const float* __restrict__ out_b,
               float* __restrict__ out) {
    __shared__ float lds_y[32 * LDSX];
    const int blk = blockIdx.x;
    const int b   = blk >> 6;
    const int s0  = (blk & 63) << 5;
    const int t   = threadIdx.x;

    const float* src = ytot + ((size_t)(b * SEQ + s0)) * DM;
    #pragma unroll
    for (int e = 0; e < 8; ++e) {
        int idx = t + e * 256;
        int row = idx >> 6, c4 = idx & 63;
        stage_b128(src + (size_t)row * DM + c4 * 4, &lds_y[row * LDSX + c4 * 4]);
    }
    stage_wait();
    __syncthreads();

    const int lane = t & 31, wave = t >> 5;
    const int r = lane & 15, hi = lane >> 4;
    const int n0 = wave << 5;

    v8f a00 = {}, a01 = {}, a10 = {}, a11 = {};
    for (int k = 0; k < DM; k += 4) {
        const int kk = k + 2 * hi;
        v2f fa0 = *(const v2f*)&lds_y[r * LDSX + kk];
        v2f fa1 = *(const v2f*)&lds_y[(16 + r) * LDSX + kk];
        const float* w0 = out_w + (size_t)(n0 + r) * DM + kk;
        const float* w1 = w0 + (size_t)16 * DM;
        v2f fb0; fb0.x = w0[0]; fb0.y = w0[1];
        v2f fb1; fb1.x = w1[0]; fb1.y = w1[1];
        a00 = wmma4_f32(fa0, fb0, a00);
        a10 = wmma4_f32(fa1, fb0, a10);
        a01 = wmma4_f32(fa0, fb1, a01);
        a11 = wmma4_f32(fa1, fb1, a11);
    }

    float* go = out + ((size_t)(b * SEQ + s0)) * DM;
    const int n = n0 + r, n2 = n + 16;
    const float bi = out_b[n], bi2 = out_b[n2];
    #pragma unroll
    for (int rr = 0; rr < 8; ++rr) {
        int m = rr + 8 * hi;
        go[(size_t)m * DM + n ]        = a00[rr] + bi;
        go[(size_t)m * DM + n2]        = a01[rr] + bi2;
        go[(size_t)(m+16) * DM + n ]   = a10[rr] + bi;
        go[(size_t)(m+16) * DM + n2]   = a11[rr] + bi2;
    }
}

// ---------------------------------------------------------------------------
extern "C" void kernel_launch(void* const* d_in, const int* in_sizes, int n_in,
                              void* d_out, int out_size, void* d_ws, size_t ws_size,
                              hipStream_t stream) {
    const float* input  = (const float*)d_in[0];   // (8,2048,512)
    const float* A_log  = (const float*)d_in[1];   // (256,16)
    const float* B_ssm  = (const float*)d_in[2];
    const float* C_ssm  = (const float*)d_in[3];
    const float* D_ssm  = (const float*)d_in[4];   // (256,)
    const float* in_w   = (const float*)d_in[5];   // (256,256)
    const float* in_b   = (const float*)d_in[6];
    const float* out_w  = (const float*)d_in[7];
    const float* out_b  = (const float*)d_in[8];
    const float* gate_w = (const float*)d_in[9];
    const float* gate_b = (const float*)d_in[10];
    const float* conv_w = (const float*)d_in[11];  // (256,1,3)
    const float* conv_b = (const float*)d_in[12];

    float* out   = (float*)d_out;
    const size_t planeBytes = (size_t)BSZ * SEQ * DM * sizeof(float);   // 16 MB
    float* xproj = (float*)d_ws;
    float* xg    = (float*)((char*)d_ws + planeBytes);                  // xg -> ytot in place
    float* cs    = (float*)((char*)d_ws + 2 * planeBytes);              // 4 MB chunk states

    const int nblk = BSZ * (SEQ / 32);   // 512

    k_proj_gate<<<nblk, 256, 0, stream>>>(input, in_w, in_b, gate_w, gate_b, xproj, xg);
    k_scan_local  <<<BSZ * NCHUNK, DM, 0, stream>>>(xg, A_log, B_ssm, cs);
    k_scan_combine<<<BSZ,          DM, 0, stream>>>(A_log, cs);
    k_scan_final  <<<BSZ * NCHUNK, DM, 0, stream>>>(xproj, xg, A_log, B_ssm, C_ssm,
                                                    D_ssm, conv_w, conv_b, cs);
    k_outproj<<<nblk, 256, 0, stream>>>(xg, out_w, out_b, out);
}